// MeasureDistance_2087354106222
// MI455X (gfx1250) — compile-verified
//
#include <hip/hip_runtime.h>
#include <hip/hip_bf16.h>
#include <stdint.h>

#define USE_ASYNC_STAGE 1

typedef float v2f __attribute__((ext_vector_type(2)));
typedef float v8f __attribute__((ext_vector_type(8)));

namespace {
constexpr int kB = 8;
constexpr int kN = 2048;   // both L and K
constexpr int kD = 64;
constexpr float kTol = 1e-6f;
constexpr float kSqrtEps = 1e-8f;
constexpr float kLog2e = 1.4426950408889634f;
constexpr float kLn2   = 0.6931471805599453f;
constexpr int kTileStride = 68;  // 16B-aligned rows (68*4=272 = 0 mod 16), bank-safe
}

__device__ __forceinline__ float neg_inf() { return -__builtin_huge_valf(); }

// ---------------------------------------------------------------------------
// Squared norms of each point: out[b*N+i] = sum_d pts[b,i,d]^2
// ---------------------------------------------------------------------------
__global__ void norms_kernel(const float* __restrict__ pts, float* __restrict__ out) {
  int idx = blockIdx.x * blockDim.x + threadIdx.x;  // 0 .. B*N-1
  const float4* r = (const float4*)(pts + (size_t)idx * kD);
  float s = 0.f;
#pragma unroll
  for (int k = 0; k < kD / 4; ++k) {
    float4 v = r[k];
    s += v.x * v.x + v.y * v.y + v.z * v.z + v.w * v.w;
  }
  out[idx] = s;
}

// ---------------------------------------------------------------------------
// Zero the 4 potential arrays (contiguous, 4*B*N floats) and set flags = 1
// ---------------------------------------------------------------------------
__global__ void init_kernel(float* __restrict__ pots, int* __restrict__ flags) {
  int idx = blockIdx.x * blockDim.x + threadIdx.x;
  if (idx < 4 * kB * kN) pots[idx] = 0.f;
  if (idx < 4) flags[idx] = 1;
}

// ---------------------------------------------------------------------------
// Fused extrapolate:
//   out[b,r] = -LSE_c( -sqrt(max(rowN[r]+colN[c]-2*dot(rowP[r],colP[c]),0)+eps)
//              + colW[c] + pot[c] )
// Grid: B*32 blocks x 128 threads (4 waves). Each wave owns one 16-row strip;
// the 4 waves share async-staged (global_load_async_to_lds_b128) 16x64 column
// tiles in LDS. Gram dot products use V_WMMA_F32_16X16X4_F32 (K=64, 16 steps).
// The online log-sum-exp runs in log2 domain: bare v_exp_f32 / v_log_f32.
// ---------------------------------------------------------------------------
__global__ void extrap_kernel(const float* __restrict__ rowP,
                              const float* __restrict__ colP,
                              const float* __restrict__ rowN,
                              const float* __restrict__ colN,
                              const float* __restrict__ colW,
                              const float* __restrict__ pot,
                              float* __restrict__ outv) {
  __shared__ float ltile[16 * kTileStride];
  __shared__ float lcn[16];
  __shared__ float lca[16];

  const int tid  = threadIdx.x;     // 0..127
  const int w    = tid >> 5;        // wave 0..3
  const int l    = tid & 31;        // lane
  const int ln   = l & 15;          // lane within half
  const int half = (l >= 16) ? 1 : 0;
  const int kb   = half * 2;        // K sub-offset inside each K=4 chunk

  const int b     = blockIdx.x >> 5;               // batch (32 blocks/batch)
  const int strip = ((blockIdx.x & 31) << 2) | w;  // 0..127
  const int m0    = strip << 4;                    // first row of strip

  // --- A fragments: rows m0..m0+15, full K=64, held in registers ---
  const float* rp = rowP + ((size_t)b * kN + m0 + ln) * kD + kb;
  v2f afrag[16];
#pragma unroll
  for (int kc = 0; kc < 16; ++kc) afrag[kc] = *(const v2f*)(rp + kc * 4);

  float rn[8];
#pragma unroll
  for (int j = 0; j < 8; ++j) rn[j] = rowN[(size_t)b * kN + m0 + half * 8 + j];

  float mj[8], sj[8];  // log2-domain running max / scaled sum
#pragma unroll
  for (int j = 0; j < 8; ++j) { mj[j] = neg_inf(); sj[j] = 0.f; }

  const size_t cb = (size_t)b * kN;

  for (int n0 = 0; n0 < kN; n0 += 16) {
    __syncthreads();  // protect LDS from previous tile's consumers
    // --- stage 16x64 column tile (2 x b128 per thread) + per-column scalars ---
    {
#pragma unroll
      for (int part = 0; part < 2; ++part) {
        int flat = tid + part * 128;       // 0..255
        int row  = flat >> 4;              // 0..15
        int off  = (flat & 15) << 2;       // 0..60, 16B-aligned chunks
        const float* gsrc = colP + (cb + n0 + row) * (size_t)kD + off;
        float* dst = ltile + row * kTileStride + off;
#if USE_ASYNC_STAGE
        unsigned int laddr = (unsigned int)(uintptr_t)dst;
        asm volatile("global_load_async_to_lds_b128 %0, %1, off"
                     :: "v"(laddr), "v"((unsigned long long)(uintptr_t)gsrc)
                     : "memory");
#else
        float4 v = *(const float4*)gsrc;
        dst[0] = v.x; dst[1] = v.y; dst[2] = v.z; dst[3] = v.w;
#endif
      }
      if (tid < 16) {
        lcn[tid] = colN[cb + n0 + tid];
      } else if (tid < 32) {
        int c = tid - 16;
        lca[c] = (colW[cb + n0 + c] + pot[cb + n0 + c]) * kLog2e;  // log2 domain
      }
#if USE_ASYNC_STAGE
      asm volatile("s_wait_asynccnt 0x0" ::: "memory");
#endif
    }
    __syncthreads();

    // --- Gram tile via WMMA f32 16x16x4, accumulate over K=64 ---
    v8f acc = {0.f, 0.f, 0.f, 0.f, 0.f, 0.f, 0.f, 0.f};
    const float* bp = ltile + ln * kTileStride + kb;
#pragma unroll
    for (int kc = 0; kc < 16; ++kc) {
      v2f bfrag = *(const v2f*)(bp + kc * 4);
      acc = __builtin_amdgcn_wmma_f32_16x16x4_f32(
          false, afrag[kc], false, bfrag, (short)0, acc, false, false);
    }

    // --- energy kernel + online log2-sum-exp2 per row ---
    const float cn  = lcn[ln];
    const float ca2 = lca[ln];  // (colW + pot) * log2(e)
#pragma unroll
    for (int j = 0; j < 8; ++j) {
      float dist = rn[j] + cn - 2.0f * acc[j];
      dist = fmaxf(dist, 0.0f) + kSqrtEps;
      float rt = __builtin_amdgcn_sqrtf(dist);     // raw v_sqrt_f32
      float v2 = fmaf(rt, -kLog2e, ca2);           // (ca - sqrt)*log2e
      float nm = fmaxf(mj[j], v2);
      sj[j] = fmaf(sj[j], __builtin_amdgcn_exp2f(mj[j] - nm),
                   __builtin_amdgcn_exp2f(v2 - nm));
      mj[j] = nm;
    }
  }

  // --- cross-lane LSE merge within each 16-lane half (columns) ---
#pragma unroll
  for (int off = 8; off >= 1; off >>= 1) {
#pragma unroll
    for (int j = 0; j < 8; ++j) {
      float mo = __shfl_xor(mj[j], off, 32);
      float so = __shfl_xor(sj[j], off, 32);
      float nm = fmaxf(mj[j], mo);
      sj[j] = fmaf(sj[j], __builtin_amdgcn_exp2f(mj[j] - nm),
                   so * __builtin_amdgcn_exp2f(mo - nm));
      mj[j] = nm;
    }
  }

  if (ln == 0) {
    size_t rbase = (size_t)b * kN + m0 + half * 8;
#pragma unroll
    for (int j = 0; j < 8; ++j) {
      // natural-log result: -(m2 + log2(s)) * ln2
      outv[rbase + j] = -kLn2 * (mj[j] + __builtin_amdgcn_logf(sj[j]));
    }
  }
}

// ---------------------------------------------------------------------------
// Per-batch max/min of (new - pot) for the gap, and damped pot update gated
// by the carry flag. Grid: B blocks x 256 threads.
// ---------------------------------------------------------------------------
__global__ void reduce_update_kernel(const float* __restrict__ newv,
                                     float* __restrict__ pot,
                                     const int* __restrict__ flags, int fidx,
                                     float* __restrict__ redMax,
                                     float* __restrict__ redMin) {
  __shared__ float smx[8], smn[8];
  const int b = blockIdx.x;
  const int tid = threadIdx.x;
  const int flag = flags[fidx];
  const size_t base = (size_t)b * kN;
  float mx = neg_inf(), mn = -neg_inf();
  for (int i = tid; i < kN; i += 256) {
    float pv = pot[base + i];
    float nv = newv[base + i];
    float d = nv - pv;
    mx = fmaxf(mx, d);
    mn = fminf(mn, d);
    if (flag) pot[base + i] = 0.5f * pv + 0.5f * nv;
  }
#pragma unroll
  for (int off = 16; off >= 1; off >>= 1) {
    mx = fmaxf(mx, __shfl_xor(mx, off, 32));
    mn = fminf(mn, __shfl_xor(mn, off, 32));
  }
  if ((tid & 31) == 0) { smx[tid >> 5] = mx; smn[tid >> 5] = mn; }
  __syncthreads();
  if (tid == 0) {
    float MX = smx[0], MN = smn[0];
    for (int i = 1; i < 8; ++i) { MX = fmaxf(MX, smx[i]); MN = fminf(MN, smn[i]); }
    redMax[fidx * kB + b] = MX;
    redMin[fidx * kB + b] = MN;
  }
}

// ---------------------------------------------------------------------------
// End-of-iteration flag update with the reference's yx/xy sync coupling.
// ---------------------------------------------------------------------------
__global__ void flag_combine_kernel(const float* __restrict__ redMax,
                                    const float* __restrict__ redMin,
                                    int* __restrict__ flags) {
  if (threadIdx.x == 0 && blockIdx.x == 0) {
    float g[4];
    for (int f = 0; f < 4; ++f) {
      float s = 0.f;
      for (int b = 0; b < kB; ++b) s += redMax[f * kB + b] - redMin[f * kB + b];
      g[f] = s / (float)kB;
    }
    int rxx = flags[0] && (g[0] >= kTol);
    int ryx = flags[1] && (g[1] >= kTol);
    int rxy = flags[2] && (g[2] >= kTol);
    int ryy = flags[3] && (g[3] >= kTol);
    int syn = ryx || rxy;
    flags[0] = rxx; flags[1] = syn; flags[2] = syn; flags[3] = ryy;
  }
}

// ---------------------------------------------------------------------------
// out[b] = sum_l (exy-exx)*exp(a) + sum_k (eyx-eyy)*exp(bw)
// ---------------------------------------------------------------------------
__global__ void final_kernel(const float* __restrict__ exx, const float* __restrict__ exy,
                             const float* __restrict__ eyx, const float* __restrict__ eyy,
                             const float* __restrict__ aw, const float* __restrict__ bw,
                             float* __restrict__ out) {
  __shared__ float sp[8];
  const int b = blockIdx.x;
  const int tid = threadIdx.x;
  const size_t base = (size_t)b * kN;
  float s = 0.f;
  for (int i = tid; i < kN; i += 256) {
    s += (exy[base + i] - exx[base + i]) * __expf(aw[base + i]);
    s += (eyx[base + i] - eyy[base + i]) * __expf(bw[base + i]);
  }
#pragma unroll
  for (int off = 16; off >= 1; off >>= 1) s += __shfl_xor(s, off, 32);
  if ((tid & 31) == 0) sp[tid >> 5] = s;
  __syncthreads();
  if (tid == 0) {
    float t = 0.f;
    for (int i = 0; i < 8; ++i) t += sp[i];
    out[b] = t;
  }
}

// ---------------------------------------------------------------------------
extern "C" void kernel_launch(void* const* d_in, const int* in_sizes, int n_in,
                              void* d_out, int out_size, void* d_ws, size_t ws_size,
                              hipStream_t stream) {
  (void)in_sizes; (void)n_in; (void)out_size; (void)ws_size;
  const float* x  = (const float*)d_in[0];
  const float* a  = (const float*)d_in[1];
  const float* y  = (const float*)d_in[2];
  const float* bw = (const float*)d_in[3];
  float* out = (float*)d_out;

  float* w = (float*)d_ws;
  const size_t S = (size_t)kB * kN;  // 16384
  float* nx  = w + 0 * S;
  float* ny  = w + 1 * S;
  float* pxx = w + 2 * S;   // 4 pot arrays contiguous: pxx,pxy,pyx,pyy
  float* pxy = w + 3 * S;
  float* pyx = w + 4 * S;
  float* pyy = w + 5 * S;
  float* nb  = w + 6 * S;   // "new" scratch
  float* exx = w + 7 * S;
  float* eyx = w + 8 * S;
  float* exy = w + 9 * S;
  float* eyy = w + 10 * S;
  float* rmx = w + 11 * S;
  float* rmn = rmx + 4 * kB;
  int*   flg = (int*)(rmn + 4 * kB);

  init_kernel<<<dim3((4 * kB * kN) / 256), dim3(256), 0, stream>>>(pxx, flg);
  norms_kernel<<<dim3((kB * kN) / 256), dim3(256), 0, stream>>>(x, nx);
  norms_kernel<<<dim3((kB * kN) / 256), dim3(256), 0, stream>>>(y, ny);

  const dim3 eg(kB * 32), eb(128);
  for (int it = 0; it < 10; ++it) {
    // xx: new = extrap(pxx, Kxx) -> update pxx
    extrap_kernel<<<eg, eb, 0, stream>>>(x, x, nx, nx, a, pxx, nb);
    reduce_update_kernel<<<dim3(kB), dim3(256), 0, stream>>>(nb, pxx, flg, 0, rmx, rmn);
    // yx: new = extrap(pxy, Kyx) -> update pyx
    extrap_kernel<<<eg, eb, 0, stream>>>(y, x, ny, nx, a, pxy, nb);
    reduce_update_kernel<<<dim3(kB), dim3(256), 0, stream>>>(nb, pyx, flg, 1, rmx, rmn);
    // xy: new = extrap(pyx_fresh, Kxy) -> update pxy
    extrap_kernel<<<eg, eb, 0, stream>>>(x, y, nx, ny, bw, pyx, nb);
    reduce_update_kernel<<<dim3(kB), dim3(256), 0, stream>>>(nb, pxy, flg, 2, rmx, rmn);
    // yy: new = extrap(pyy, Kyy) -> update pyy
    extrap_kernel<<<eg, eb, 0, stream>>>(y, y, ny, ny, bw, pyy, nb);
    reduce_update_kernel<<<dim3(kB), dim3(256), 0, stream>>>(nb, pyy, flg, 3, rmx, rmn);
    flag_combine_kernel<<<dim3(1), dim3(32), 0, stream>>>(rmx, rmn, flg);
  }

  // final extrapolation of every direction
  extrap_kernel<<<eg, eb, 0, stream>>>(x, x, nx, nx, a, pxx, exx);
  extrap_kernel<<<eg, eb, 0, stream>>>(y, x, ny, nx, a, pxy, eyx);
  extrap_kernel<<<eg, eb, 0, stream>>>(x, y, nx, ny, bw, pyx, exy);
  extrap_kernel<<<eg, eb, 0, stream>>>(y, y, ny, ny, bw, pyy, eyy);
  final_kernel<<<dim3(kB), dim3(256), 0, stream>>>(exx, exy, eyx, eyy, a, bw, out);
}